// NOENet_73899207295066
// MI455X (gfx1250) — compile-verified
//
#include <hip/hip_runtime.h>

#define B_TOT 8192
#define T_LEN 2048
#define TM1   (T_LEN - 1)
#define HID   20
#define NROW  40            // 20 z-rows + 20 r-rows
#define NWG   16
#define TPB   512
#define NWAVE (TPB / 32)
#define EPSV  1e-5f

static_assert(NWG * TPB == B_TOT, "grid must cover batch exactly");

typedef float v2f __attribute__((ext_vector_type(2)));
typedef float v8f __attribute__((ext_vector_type(8)));

// ---------------------------------------------------------------- init
__global__ void ws_init_kernel(unsigned* counter) { *counter = 0u; }

// ---------------------------------------------------- per-column x stats
// sx[t] = sum_b x[b,t]; sxx[t] = sum_b x[b,t]^2; sxy[t] = sum_b x[b,t]*x[b,t+1]
__global__ __launch_bounds__(256) void colstats_kernel(
    const float* __restrict__ x,
    float* __restrict__ sx, float* __restrict__ sxx, float* __restrict__ sxy)
{
    __shared__ float r0[8][33], r1[8][33], r2[8][33];
    const int cl  = threadIdx.x & 31;
    const int row = threadIdx.x >> 5;
    const int c   = blockIdx.x * 32 + cl;
    const bool has_next = (c + 1) < T_LEN;
    float a0 = 0.f, a1 = 0.f, a2 = 0.f;
    for (int bb = row; bb < B_TOT; bb += 8) {
        size_t base = (size_t)bb * T_LEN + c;
        float v = x[base];
        a0 += v; a1 += v * v;
        if (has_next) a2 += v * x[base + 1];
    }
    r0[row][cl] = a0; r1[row][cl] = a1; r2[row][cl] = a2;
    __syncthreads();
    if (row == 0) {
        float s0 = 0.f, s1 = 0.f, s2 = 0.f;
        #pragma unroll
        for (int k = 0; k < 8; ++k) { s0 += r0[k][cl]; s1 += r1[k][cl]; s2 += r2[k][cl]; }
        sx[c] = s0; sxx[c] = s1; sxy[c] = s2;
    }
}

// ---------------------------------------------------------- main recurrence
__global__ __launch_bounds__(TPB) void noenet_kernel(
    const float* __restrict__ x,
    const float* __restrict__ W1,
    const float* __restrict__ g1,  const float* __restrict__ bt1,
    const float* __restrict__ W2,  const float* __restrict__ b2,
    const float* __restrict__ RW1,
    const float* __restrict__ rg1, const float* __restrict__ rbt1,
    const float* __restrict__ RW2, const float* __restrict__ rb2,
    float* __restrict__ out,
    unsigned* __restrict__ counter,
    float* __restrict__ partial,              // [2][NWG][16]
    const float* __restrict__ sx, const float* __restrict__ sxx,
    const float* __restrict__ sxy)
{
    __shared__ float inp_s[4][TPB + 16];      // SoA staging of [u0,u1,h0,h1]
    __shared__ float Wp[48][4];               // folded weights (rows 40..47 = 0 pad)
    __shared__ float bp[48];                  // folded bias
    __shared__ float wpart[NWAVE][16];        // per-wave partial stats
    __shared__ float stats[16];               // mu[4] + upper-tri M[10]

    const int tid  = threadIdx.x;
    const int wg   = blockIdx.x;
    const int lane = tid & 31;
    const int wave = tid >> 5;
    const int n    = lane & 15;
    const int kh   = lane >> 4;               // 0/1 half of wave
    const int b    = wg * TPB + tid;          // owned batch element

    if (tid < 8) {                            // zero padding rows 40..47 once
        int j = NROW + tid;
        bp[j] = 0.f; Wp[j][0] = 0.f; Wp[j][1] = 0.f; Wp[j][2] = 0.f; Wp[j][3] = 0.f;
    }

    // per-lane epilogue weights for rows j = 16*tm + 8*kh + v (constant over t)
    float wy[3][8], wr0[3][8], wr1[3][8];
    #pragma unroll
    for (int tm = 0; tm < 3; ++tm)
        #pragma unroll
        for (int v = 0; v < 8; ++v) {
            const int j = 16 * tm + 8 * kh + v;
            const bool zrow = (j < HID);
            const bool rrow = (j >= HID) && (j < NROW);
            wy [tm][v] = zrow ? W2[j]         : 0.f;   // 20->1 head
            wr0[tm][v] = rrow ? RW2[j - HID]  : 0.f;   // RW2[0][j-20]
            wr1[tm][v] = rrow ? RW2[j]        : 0.f;   // RW2[1][j-20]
        }
    const float b2v  = b2[0];
    const float rb20 = rb2[0], rb21 = rb2[1];

    float h0 = 0.f, h1 = 0.f;
    float u0 = x[(size_t)b * T_LEN];

    for (int t = 0; t < TM1; ++t) {
        const int slot = t & 1;
        float u1 = x[(size_t)b * T_LEN + t + 1];

        // stage inp for WMMA B operand (wave-local region only)
        inp_s[0][tid] = u0; inp_s[1][tid] = u1; inp_s[2][tid] = h0; inp_s[3][tid] = h1;

        // 9 h-coupled batch statistics (rest precomputed from x)
        float p[9];
        p[0] = h0;      p[1] = h1;
        p[2] = u0 * h0; p[3] = u0 * h1; p[4] = u1 * h0; p[5] = u1 * h1;
        p[6] = h0 * h0; p[7] = h0 * h1; p[8] = h1 * h1;
        #pragma unroll
        for (int s = 0; s < 9; ++s) {
            #pragma unroll
            for (int off = 16; off > 0; off >>= 1)
                p[s] += __shfl_xor(p[s], off, 32);
        }
        if (lane == 0) {
            #pragma unroll
            for (int s = 0; s < 9; ++s) wpart[wave][s] = p[s];
        }
        __syncthreads();                                           // (a)

        if (tid < 9) {
            float acc = 0.f;
            #pragma unroll
            for (int w = 0; w < NWAVE; ++w) acc += wpart[w][tid];
            __hip_atomic_store(&partial[(slot * NWG + wg) * 16 + tid], acc,
                               __ATOMIC_RELEASE, __HIP_MEMORY_SCOPE_AGENT);
        }
        __threadfence();
        __syncthreads();                                           // (b)

        if (tid == 0) {                                            // global barrier
            __hip_atomic_fetch_add(counter, 1u, __ATOMIC_ACQ_REL, __HIP_MEMORY_SCOPE_AGENT);
            const unsigned target = (unsigned)(NWG * (t + 1));
            while (__hip_atomic_load(counter, __ATOMIC_ACQUIRE,
                                     __HIP_MEMORY_SCOPE_AGENT) < target)
                __builtin_amdgcn_s_sleep(2);
        }
        __syncthreads();                                           // (c)

        const float invB = 1.f / (float)B_TOT;
        if (tid < 9) {                                             // combine h-stats
            float acc = 0.f;
            for (int gi = 0; gi < NWG; ++gi)
                acc += __hip_atomic_load(&partial[(slot * NWG + gi) * 16 + tid],
                                         __ATOMIC_ACQUIRE, __HIP_MEMORY_SCOPE_AGENT);
            const int idx = (tid < 2) ? (tid + 2) : ((tid < 4) ? (tid + 4) : (tid + 5));
            stats[idx] = acc * invB;
        } else if (tid < 14) {                                     // x-only stats
            float v; int idx;
            if      (tid ==  9) { v = sx [t];     idx = 0; }
            else if (tid == 10) { v = sx [t + 1]; idx = 1; }
            else if (tid == 11) { v = sxx[t];     idx = 4; }
            else if (tid == 12) { v = sxy[t];     idx = 5; }
            else                { v = sxx[t + 1]; idx = 8; }
            stats[idx] = v * invB;
        }
        __syncthreads();                                           // (d)

        if (tid < NROW) {                                          // fold BN into W',b'
            const int j = tid;
            const float* wr = (j < HID) ? (W1 + 4 * j) : (RW1 + 4 * (j - HID));
            const float gam = (j < HID) ? g1[j]  : rg1[j - HID];
            const float bta = (j < HID) ? bt1[j] : rbt1[j - HID];
            const float w0 = wr[0], w1 = wr[1], w2v = wr[2], w3 = wr[3];
            const float m0 = stats[0], m1 = stats[1], m2 = stats[2], m3 = stats[3];
            const float C00 = stats[4]  - m0 * m0, C01 = stats[5]  - m0 * m1;
            const float C02 = stats[6]  - m0 * m2, C03 = stats[7]  - m0 * m3;
            const float C11 = stats[8]  - m1 * m1, C12 = stats[9]  - m1 * m2;
            const float C13 = stats[10] - m1 * m3, C22 = stats[11] - m2 * m2;
            const float C23 = stats[12] - m2 * m3, C33 = stats[13] - m3 * m3;
            float var = w0*w0*C00 + w1*w1*C11 + w2v*w2v*C22 + w3*w3*C33
                      + 2.f * (w0*w1*C01 + w0*w2v*C02 + w0*w3*C03
                             + w1*w2v*C12 + w1*w3*C13 + w2v*w3*C23);
            var = fmaxf(var, 0.f);
            const float dotmu = w0*m0 + w1*m1 + w2v*m2 + w3*m3;
            const float sc = gam * rsqrtf(var + EPSV);
            Wp[j][0] = sc * w0; Wp[j][1] = sc * w1; Wp[j][2] = sc * w2v; Wp[j][3] = sc * w3;
            bp[j] = bta - sc * dotmu;
        }
        __syncthreads();                                           // (e)

        // ---- WMMA phase: D = W'(48x4) x inp^T(4x16) + bias, per 16-col half ----
        const int wb = wave * 32;
        const int k0 = 2 * kh, k1 = 2 * kh + 1;
        v2f A0 = { Wp[n][k0],      Wp[n][k1]      };
        v2f A1 = { Wp[16 + n][k0], Wp[16 + n][k1] };
        v2f A2 = { Wp[32 + n][k0], Wp[32 + n][k1] };
        v8f C0, C1, C2;                     // bias broadcast along N (depends on M only)
        #pragma unroll
        for (int v = 0; v < 8; ++v) {
            C0[v] = bp[ 0 + 8 * kh + v];
            C1[v] = bp[16 + 8 * kh + v];
            C2[v] = bp[32 + 8 * kh + v];
        }

        float yp[2], hp0[2], hp1[2];
        #pragma unroll
        for (int ch = 0; ch < 2; ++ch) {
            const int col = wb + 16 * ch + n;
            v2f Bv = { inp_s[k0][col], inp_s[k1][col] };
            v8f d0 = __builtin_amdgcn_wmma_f32_16x16x4_f32(false, A0, false, Bv, (short)0, C0, false, false);
            v8f d1 = __builtin_amdgcn_wmma_f32_16x16x4_f32(false, A1, false, Bv, (short)0, C1, false, false);
            v8f d2 = __builtin_amdgcn_wmma_f32_16x16x4_f32(false, A2, false, Bv, (short)0, C2, false, false);
            float ay = 0.f, a0s = 0.f, a1s = 0.f;
            #pragma unroll
            for (int v = 0; v < 8; ++v) {
                float e0 = d0[v]; e0 = fmaxf(e0, 0.2f * e0);   // leaky relu
                float e1 = d1[v]; e1 = fmaxf(e1, 0.2f * e1);
                float e2 = d2[v]; e2 = fmaxf(e2, 0.2f * e2);
                ay  += e0 * wy [0][v] + e1 * wy [1][v] + e2 * wy [2][v];
                a0s += e0 * wr0[0][v] + e1 * wr0[1][v] + e2 * wr0[2][v];
                a1s += e0 * wr1[0][v] + e1 * wr1[1][v] + e2 * wr1[2][v];
            }
            yp[ch] = ay; hp0[ch] = a0s; hp1[ch] = a1s;
        }
        // combine M=0..7 (lane n) with M=8..15 (lane n+16) halves
        #pragma unroll
        for (int ch = 0; ch < 2; ++ch) {
            yp [ch] += __shfl_xor(yp [ch], 16, 32);
            hp0[ch] += __shfl_xor(hp0[ch], 16, 32);
            hp1[ch] += __shfl_xor(hp1[ch], 16, 32);
        }
        // owned element corresponds to column-half == kh
        out[(size_t)b * TM1 + t] = yp[kh] + b2v;
        h0 = hp0[kh] + rb20;
        h1 = hp1[kh] + rb21;
        u0 = u1;

        if (((t + 1) & 31) == 0 && (t + 64) < T_LEN)
            __builtin_prefetch(&x[(size_t)b * T_LEN + t + 64], 0, 1);
    }
}

// ---------------------------------------------------------------- launch
extern "C" void kernel_launch(void* const* d_in, const int* in_sizes, int n_in,
                              void* d_out, int out_size, void* d_ws, size_t ws_size,
                              hipStream_t stream) {
    (void)in_sizes; (void)n_in; (void)out_size; (void)ws_size;
    const float* x    = (const float*)d_in[0];
    const float* W1   = (const float*)d_in[1];
    // d_in[2] = b1  (cancels inside BatchNorm: z - mean(z) removes it)
    const float* g1   = (const float*)d_in[3];
    const float* bt1  = (const float*)d_in[4];
    const float* W2   = (const float*)d_in[5];
    const float* b2   = (const float*)d_in[6];
    const float* RW1  = (const float*)d_in[7];
    // d_in[8] = rb1 (cancels)
    const float* rg1  = (const float*)d_in[9];
    const float* rbt1 = (const float*)d_in[10];
    const float* RW2  = (const float*)d_in[11];
    const float* rb2  = (const float*)d_in[12];
    float* out = (float*)d_out;

    unsigned char* ws = (unsigned char*)d_ws;
    unsigned* counter = (unsigned*)ws;                 // 4 B
    float* partial    = (float*)(ws + 256);            // 2*NWG*16 floats = 2 KB
    float* sx         = (float*)(ws + 4096);           // T floats
    float* sxx        = sx  + T_LEN;
    float* sxy        = sxx + T_LEN;                   // total ws use < 32 KB

    ws_init_kernel<<<1, 1, 0, stream>>>(counter);
    colstats_kernel<<<T_LEN / 32, 256, 0, stream>>>(x, sx, sxx, sxy);
    noenet_kernel<<<NWG, TPB, 0, stream>>>(x, W1, g1, bt1, W2, b2,
                                           RW1, rg1, rbt1, RW2, rb2,
                                           out, counter, partial, sx, sxx, sxy);
}